// MapleAttention_2456721293597
// MI455X (gfx1250) — compile-verified
//
#include <hip/hip_runtime.h>

// ---------------------------------------------------------------- constants
#define HQ      32
#define HKV     8
#define GRP     4
#define HD      128
#define ROPE_D  64
#define WINDOW  1024
#define NB      16
#define HID     4096
#define MAXCTX  4096
#define SK      8        // split-K for GEMMs
#define ASPLIT  8        // flash-decode splits
#define AWAVES  8        // waves per attention block
#define QKV_COLS 6144    // 4096 q + 1024 k + 1024 v

typedef __attribute__((ext_vector_type(2))) float v2f;
typedef __attribute__((ext_vector_type(8))) float v8f;
typedef int v4i_ld __attribute__((vector_size(4 * sizeof(int))));

#if defined(__gfx1250__) && __has_builtin(__builtin_amdgcn_global_load_async_to_lds_b128)
#define HAVE_ASYNC_LDS 1
typedef __attribute__((address_space(1))) v4i_ld* async_gptr;
typedef __attribute__((address_space(3))) v4i_ld* async_lptr;
#else
#define HAVE_ASYNC_LDS 0
#endif

// ---------------------------------------------------------------------------
// 16xN fp32 GEMM, C[16,N] = A[16,K] * W[N,K]^T, via V_WMMA_F32_16X16X4_F32.
// One wave per 16-column tile per K-chunk.  ISA f32 operand layouts:
//   A 16x4 : lanes 0-15 hold (K=0,K=1), lanes 16-31 hold (K=2,K=3), M=lane&15
//   B 4x16 : lanes 0-15 hold (K=0,K=1), lanes 16-31 hold (K=2,K=3), N=lane&15
//   C 16x16: VGPR r = row r (lanes<16) / row 8+r (lanes>=16), N=lane&15
// Partials: part[kc][m][colOff + col].  Single flat unrolled loop so the
// compiler emits one software-pipelined body (clause'd b64 loads + staggered
// s_wait_loadcnt) with no serialized remainder path.
// ---------------------------------------------------------------------------
__global__ __launch_bounds__(32)
void gemm16_wmma(const float* __restrict__ A, const float* __restrict__ W,
                 float* __restrict__ part, int K, int kchunk,
                 int outStride, int colOff, int sliceSz) {
  const int tile = blockIdx.x;
  const int kc   = blockIdx.y;
  const int lane = threadIdx.x;
  const int col0 = tile << 4;
  const int rlo  = lane & 15;
  const int koff = (lane >> 4) << 1;          // 0 or 2

  const float* ap = A + (size_t)rlo * K + (size_t)kc * kchunk + koff;
  const float* bp = W + (size_t)(col0 + rlo) * K + (size_t)kc * kchunk + koff;

  v8f acc = {0.f, 0.f, 0.f, 0.f, 0.f, 0.f, 0.f, 0.f};
  #pragma unroll 16
  for (int k = 0; k < kchunk; k += 4) {
    v2f av = *(const v2f*)(ap + k);
    v2f bv = *(const v2f*)(bp + k);
    acc = __builtin_amdgcn_wmma_f32_16x16x4_f32(
        /*neg_a=*/false, av, /*neg_b=*/false, bv,
        /*c_mod=*/(short)0, acc, /*reuse_a=*/false, /*reuse_b=*/false);
  }

  const int m0 = (lane >> 4) << 3;
  const int n  = lane & 15;
  float* P = part + (size_t)kc * sliceSz + (size_t)(colOff + col0 + n);
  #pragma unroll
  for (int r = 0; r < 8; ++r)
    P[(size_t)(m0 + r) * outStride] = acc[r];
}

// ---------------------------------------------------------------------------
// Reduce split-K partials of the fused QKV output, then RMSNorm (q,k heads)
// and partial RoPE (first 64 dims).  One block per (token, head-row),
// head-rows: 0..31 = q heads, 32..39 = k heads, 40..47 = v heads.
// ---------------------------------------------------------------------------
__global__ __launch_bounds__(128)
void norm_rope_reduce(const float* __restrict__ part,
                      const float* __restrict__ cosd, const float* __restrict__ sind,
                      const float* __restrict__ qw, const float* __restrict__ kw,
                      float* __restrict__ qn, float* __restrict__ kn,
                      float* __restrict__ vn) {
  const int b = blockIdx.x / 48;
  const int h = blockIdx.x % 48;
  const int d = threadIdx.x;

  const size_t idx = (size_t)b * QKV_COLS + (size_t)h * HD + d;
  float x = 0.f;
  #pragma unroll
  for (int kc = 0; kc < SK; ++kc)
    x += part[(size_t)kc * (NB * QKV_COLS) + idx];

  __shared__ float red[HD];
  __shared__ float xs[HD];

  const bool isV = (h >= HQ + HKV);
  if (!isV) {
    red[d] = x * x;
    __syncthreads();
    #pragma unroll
    for (int off = 64; off > 0; off >>= 1) {
      if (d < off) red[d] += red[d + off];
      __syncthreads();
    }
    const float inv = rsqrtf(red[0] * (1.0f / HD) + 1e-6f);
    const float* w = (h < HQ) ? qw : kw;
    x = w[d] * (x * inv);
    xs[d] = x;
    __syncthreads();
    if (d < ROPE_D) {
      const float c  = cosd[b * ROPE_D + d];
      const float sn = sind[b * ROPE_D + d];
      const float other = (d < ROPE_D / 2) ? -xs[d + ROPE_D / 2] : xs[d - ROPE_D / 2];
      x = xs[d] * c + other * sn;
    }
  }

  if (h < HQ)            qn[((size_t)b * HQ  +  h)            * HD + d] = x;
  else if (h < HQ + HKV) kn[((size_t)b * HKV + (h - HQ))      * HD + d] = x;
  else                   vn[((size_t)b * HKV + (h - HQ - HKV)) * HD + d] = x;
}

// ---------------------------------------------------------------------------
// Flash-decode attention, split over the token window.
// grid = (B*HKV, ASPLIT), block = 256 (8 waves).  Each wave walks tokens
// j = jstart + split*AWAVES + wave, step ASPLIT*AWAVES, with online softmax
// per query head (G=4).  The functional KV-cache update is virtualized:
// any gathered token matching an out_cache_loc entry reads the new K/V.
// Q tile is staged into LDS via the gfx1250 async global->LDS path when the
// toolchain exposes it.  Block partials (m, l, O) go to workspace.
// ---------------------------------------------------------------------------
__global__ __launch_bounds__(256)
void attn_split(const float* __restrict__ qn, const float* __restrict__ kn,
                const float* __restrict__ vn,
                const float* __restrict__ k_pool, const float* __restrict__ v_pool,
                const int* __restrict__ req_to_token,
                const int* __restrict__ req_pool_indices,
                const int* __restrict__ seq_lens, const int* __restrict__ out_loc,
                float* __restrict__ am, float* __restrict__ al,
                float* __restrict__ ao) {
  const int bkv  = blockIdx.x;
  const int b    = bkv >> 3;
  const int kv   = bkv & 7;
  const int sp   = blockIdx.y;
  const int t    = threadIdx.x;
  const int lane = t & 31;
  const int wave = t >> 5;

  __shared__ float qs[GRP][HD];
  __shared__ int   ocl[NB];
  __shared__ float wm[AWAVES][GRP];
  __shared__ float wl[AWAVES][GRP];
  __shared__ float wo_s[AWAVES][GRP][HD];

#if HAVE_ASYNC_LDS
  // 4*128 floats = 2KB; 128 lanes x 16B async copies, tracked by ASYNCcnt
  if (t < (GRP * HD) / 4) {
    float* src = const_cast<float*>(qn + ((size_t)b * HQ + kv * GRP) * HD + t * 4);
    __builtin_amdgcn_global_load_async_to_lds_b128(
        (async_gptr)(v4i_ld*)src,
        (async_lptr)(v4i_ld*)(&qs[0][0] + t * 4),
        /*offset=*/0, /*cpol=*/0);
  }
#else
  for (int i = t; i < GRP * HD; i += 256) {
    const int g = i >> 7, d = i & (HD - 1);
    qs[g][d] = qn[((size_t)b * HQ + kv * GRP + g) * HD + d];
  }
#endif
  if (t < NB) ocl[t] = out_loc[t];
#if HAVE_ASYNC_LDS && __has_builtin(__builtin_amdgcn_s_wait_asynccnt)
  __builtin_amdgcn_s_wait_asynccnt(0);
#endif
  __syncthreads();

  float qreg[GRP][4];
  #pragma unroll
  for (int g = 0; g < GRP; ++g)
    #pragma unroll
    for (int i = 0; i < 4; ++i)
      qreg[g][i] = qs[g][lane + 32 * i];

  const int sl = seq_lens[b];
  int jstart = sl - 1 - WINDOW; if (jstart < 0) jstart = 0;
  const int jend = sl;
  const int* toks = req_to_token + (size_t)req_pool_indices[b] * MAXCTX;

  float m[GRP], l[GRP], oacc[GRP][4];
  #pragma unroll
  for (int g = 0; g < GRP; ++g) {
    m[g] = -1e30f; l[g] = 0.f;
    #pragma unroll
    for (int i = 0; i < 4; ++i) oacc[g][i] = 0.f;
  }
  const float scaling = 0.08838834764831845f;   // 1/sqrt(128)

  for (int j = jstart + sp * AWAVES + wave; j < jend; j += ASPLIT * AWAVES) {
    const int tok = toks[j];
    int sub = -1;
    #pragma unroll
    for (int i = 0; i < NB; ++i)
      sub = (tok == ocl[i] && sub < 0) ? i : sub;

    const float* kvec;
    const float* vvec;
    if (sub >= 0) {
      kvec = kn + ((size_t)sub * HKV + kv) * HD;
      vvec = vn + ((size_t)sub * HKV + kv) * HD;
    } else {
      kvec = k_pool + ((size_t)tok * HKV + kv) * HD;
      vvec = v_pool + ((size_t)tok * HKV + kv) * HD;
    }

    float kr[4], vr[4];
    #pragma unroll
    for (int i = 0; i < 4; ++i) {
      kr[i] = kvec[lane + 32 * i];
      vr[i] = vvec[lane + 32 * i];
    }

    #pragma unroll
    for (int g = 0; g < GRP; ++g) {
      float p = 0.f;
      #pragma unroll
      for (int i = 0; i < 4; ++i) p += kr[i] * qreg[g][i];
      #pragma unroll
      for (int off = 16; off > 0; off >>= 1) p += __shfl_xor(p, off, 32);
      const float sc = p * scaling;
      const float mn = fmaxf(m[g], sc);
      const float corr = __expf(m[g] - mn);
      const float e    = __expf(sc   - mn);
      l[g] = l[g] * corr + e;
      #pragma unroll
      for (int i = 0; i < 4; ++i) oacc[g][i] = oacc[g][i] * corr + e * vr[i];
      m[g] = mn;
    }
  }

  #pragma unroll
  for (int g = 0; g < GRP; ++g) {
    if (lane == 0) { wm[wave][g] = m[g]; wl[wave][g] = l[g]; }
    #pragma unroll
    for (int i = 0; i < 4; ++i) wo_s[wave][g][lane + 32 * i] = oacc[g][i];
  }
  __syncthreads();

  // combine this block's 8 waves -> one (m,l,O) partial per (b,kv,split,g)
  for (int e = t; e < GRP * HD; e += 256) {
    const int g = e >> 7, d = e & (HD - 1);
    float M = -1e30f;
    #pragma unroll
    for (int w = 0; w < AWAVES; ++w) M = fmaxf(M, wm[w][g]);
    float L = 0.f, O = 0.f;
    #pragma unroll
    for (int w = 0; w < AWAVES; ++w) {
      const float f = __expf(wm[w][g] - M);
      L += wl[w][g] * f;
      O += wo_s[w][g][d] * f;
    }
    const size_t pidx = (((size_t)bkv * ASPLIT + sp) * GRP + g);
    if (d == 0) { am[pidx] = M; al[pidx] = L; }
    ao[pidx * HD + d] = O;
  }
}

// ---------------------------------------------------------------------------
// Merge flash-decode splits (log-sum-exp) -> attn[b, h, d]
// ---------------------------------------------------------------------------
__global__ __launch_bounds__(256)
void attn_combine(const float* __restrict__ am, const float* __restrict__ al,
                  const float* __restrict__ ao, float* __restrict__ attn) {
  const int bkv = blockIdx.x;
  const int b   = bkv >> 3;
  const int kv  = bkv & 7;
  const int t   = threadIdx.x;
  for (int e = t; e < GRP * HD; e += 256) {
    const int g = e >> 7, d = e & (HD - 1);
    float M = -1e30f;
    #pragma unroll
    for (int s = 0; s < ASPLIT; ++s)
      M = fmaxf(M, am[((size_t)bkv * ASPLIT + s) * GRP + g]);
    float L = 0.f, O = 0.f;
    #pragma unroll
    for (int s = 0; s < ASPLIT; ++s) {
      const size_t pidx = ((size_t)bkv * ASPLIT + s) * GRP + g;
      const float f = __expf(am[pidx] - M);
      L += al[pidx] * f;
      O += ao[pidx * HD + d] * f;
    }
    attn[((size_t)b * HQ + kv * GRP + g) * HD + d] = O / L;
  }
}

// ---------------------------------------------------------------------------
// Reduce split-K partials of the output projection into d_out.
// ---------------------------------------------------------------------------
__global__ __launch_bounds__(256)
void reduce_out(const float* __restrict__ part, float* __restrict__ out,
                int total) {
  const int i = blockIdx.x * 256 + threadIdx.x;
  if (i < total) {
    float s = 0.f;
    #pragma unroll
    for (int kc = 0; kc < SK; ++kc) s += part[(size_t)kc * total + i];
    out[i] = s;
  }
}

// ---------------------------------------------------------------------------
extern "C" void kernel_launch(void* const* d_in, const int* in_sizes, int n_in,
                              void* d_out, int out_size, void* d_ws, size_t ws_size,
                              hipStream_t stream) {
  (void)in_sizes; (void)n_in; (void)out_size; (void)ws_size;

  const float* hid    = (const float*)d_in[0];
  const float* cosd   = (const float*)d_in[1];
  const float* sind   = (const float*)d_in[2];
  const float* k_pool = (const float*)d_in[3];
  const float* v_pool = (const float*)d_in[4];
  const float* wq     = (const float*)d_in[5];
  const float* wk     = (const float*)d_in[6];
  const float* wv     = (const float*)d_in[7];
  const float* wo     = (const float*)d_in[8];
  const float* qw     = (const float*)d_in[9];
  const float* kw     = (const float*)d_in[10];
  const int*   r2t    = (const int*)d_in[11];
  const int*   rpi    = (const int*)d_in[12];
  const int*   slen   = (const int*)d_in[13];
  const int*   oloc   = (const int*)d_in[14];
  float* out = (float*)d_out;
  float* ws  = (float*)d_ws;

  // workspace layout (floats); qkv partials and wo partials share offset 0
  // (qkv partials are dead after norm_rope_reduce, wo partials written later)
  float* qkv_part = ws;                                  // SK*16*6144 = 786432
  float* out_part = ws;                                  // SK*16*4096 = 524288
  float* qn   = ws + 786432;                             // 16*32*128 = 65536
  float* kn   = ws + 786432 + 65536;                     // 16*8*128  = 16384
  float* vn   = ws + 786432 + 65536 + 16384;             // 16384
  float* am   = ws + 786432 + 65536 + 2 * 16384;         // 16*8*8*4  = 4096
  float* al   = am + 4096;                               // 4096
  float* ao   = al + 4096;                               // 16*8*8*512 = 524288
  float* attn = ao + 524288;                             // 16*4096   = 65536

  const int kchunk = HID / SK;  // 512

  // fused QKV projection (split-K, WMMA f32 16x16x4)
  gemm16_wmma<<<dim3(HID / 16, SK), 32, 0, stream>>>(
      hid, wq, qkv_part, HID, kchunk, QKV_COLS, 0,    NB * QKV_COLS);
  gemm16_wmma<<<dim3((HKV * HD) / 16, SK), 32, 0, stream>>>(
      hid, wk, qkv_part, HID, kchunk, QKV_COLS, 4096, NB * QKV_COLS);
  gemm16_wmma<<<dim3((HKV * HD) / 16, SK), 32, 0, stream>>>(
      hid, wv, qkv_part, HID, kchunk, QKV_COLS, 5120, NB * QKV_COLS);

  // split-K reduce + RMSNorm + partial RoPE
  norm_rope_reduce<<<NB * 48, HD, 0, stream>>>(qkv_part, cosd, sind, qw, kw,
                                               qn, kn, vn);

  // flash-decode attention over the sliding window
  attn_split<<<dim3(NB * HKV, ASPLIT), 256, 0, stream>>>(
      qn, kn, vn, k_pool, v_pool, r2t, rpi, slen, oloc, am, al, ao);
  attn_combine<<<NB * HKV, 256, 0, stream>>>(am, al, ao, attn);

  // output projection (split-K, WMMA) + reduce
  gemm16_wmma<<<dim3(HID / 16, SK), 32, 0, stream>>>(
      attn, wo, out_part, HID, kchunk, HID, 0, NB * HID);
  reduce_out<<<(NB * HID + 255) / 256, 256, 0, stream>>>(out_part, out, NB * HID);
}